// HashNGramEmbedder_9337258902406
// MI455X (gfx1250) — compile-verified
//
#include <hip/hip_runtime.h>
#include <stdint.h>

// ---------------------------------------------------------------------------
// HashNGramEmbedder for gfx1250 (MI455X)
//   out[p, :] = ( main_w[tok[p]] + sum_{n=3..8} shared_w[idx_n(p)]
//               + sum_n size_w[n-3] ) / 7
// Pure gather kernel -> bandwidth bound (~522 MB => ~22us floor @ 23.3 TB/s).
// CDNA5 path: GLOBAL_LOAD_ASYNC_TO_LDS_B128 + s_wait_asynccnt double
// buffering so each wave keeps up to 14 outstanding 512B row-chunk fetches.
// Output stores are non-temporal so the write-once 64MB result does not
// evict gather-useful shared_w lines from the 192MB L2.
// ---------------------------------------------------------------------------

#define BATCH     4
#define SEQ       8192
#define DIM       512
#define NBUCKETS  500000u
#define HMASK     0x7FFFFFu   // 2^23 - 1

constexpr int TPB  = 128;  // threads per block; one float4 per thread over DIM
constexpr int PPB  = 8;    // positions pipelined per block
constexpr int NROW = 7;    // main row + 6 n-gram rows

static_assert(DIM / 4 == TPB, "one float4 lane per thread");
static_assert((BATCH * SEQ) % PPB == 0, "exact grid");
static_assert(SEQ % PPB == 0, "blocks never straddle a batch row");

#if defined(__has_builtin)
#  if __has_builtin(__builtin_amdgcn_global_load_async_to_lds_b128)
#    define HAVE_ASYNC_BUILTIN 1
#  endif
#  if __has_builtin(__builtin_amdgcn_s_wait_asynccnt)
#    define HAVE_WAIT_BUILTIN 1
#  endif
#endif

// The async builtin's params (per hipcc diagnostic) are pointers to a 16B int
// vector; AS1 prints as __device__, AS3 as __shared__ in HIP mode.
typedef int   v4i __attribute__((vector_size(4 * sizeof(int))));
typedef float v4f __attribute__((ext_vector_type(4)));  // native vec for NT store
typedef __attribute__((address_space(1))) v4i* gv4p;    // global v4i*
typedef __attribute__((address_space(3))) v4i* lv4p;    // LDS v4i*

__device__ __forceinline__ void async_ld_b128(const void* g, void* l) {
#if defined(HAVE_ASYNC_BUILTIN)
  __builtin_amdgcn_global_load_async_to_lds_b128((gv4p)g, (lv4p)l, 0, 0);
#else
  uint64_t ga = (uint64_t)g;
  uint32_t la = (uint32_t)(size_t)(__attribute__((address_space(3))) void*)l;
  asm volatile("global_load_async_to_lds_b128 %0, %1, off"
               :: "v"(la), "v"(ga)
               : "memory");
#endif
}

template <int N>
__device__ __forceinline__ void wait_async() {
#if defined(HAVE_WAIT_BUILTIN)
  __builtin_amdgcn_s_wait_asynccnt(N);
#else
  asm volatile("s_wait_asynccnt %0" :: "n"(N) : "memory");
#endif
}

// Rolling-hash row indices for one position p (uniform across the block;
// the compiler scalarizes this onto the SALU/SMEM path).
// rows[0]   = token (main_w row)
// rows[n-2] = (sum_{j<n} tok[s-j]*260^j mod 2^23) mod 500000, or 0 if s<n-1
//
// Token loads are index-clamped instead of branch-guarded: for j > s the
// loaded value only feeds hashes of windows that are force-selected to row 0
// below, so any in-bounds value is fine and the scalar branches disappear.
__device__ __forceinline__ void compute_rows(const int* __restrict__ tokens,
                                             int p, uint32_t rows[NROW]) {
  const int s    = p & (SEQ - 1);
  const int base = p - s;  // b * SEQ
  uint32_t t[8];
#pragma unroll
  for (int j = 0; j < 8; ++j) {
    int o = s - j;
    t[j] = (uint32_t)tokens[base + (o < 0 ? 0 : o)];
  }
  // 260^j mod 2^23 for j = 0..7
  const uint32_t pw[8] = {1u,       260u,     67600u,   798784u,
                          6357248u, 328704u,  1576960u, 7356416u};
  rows[0] = t[0];
  uint32_t acc = (t[0] * pw[0]) & HMASK;
  acc += (t[1] * pw[1]) & HMASK;
#pragma unroll
  for (int j = 2; j < 8; ++j) {          // n = j + 1  (3..8)
    acc += (t[j] * pw[j]) & HMASK;       // acc < 8 * 2^23, no overflow
    rows[j - 1] = (s >= j) ? ((acc & HMASK) % NBUCKETS) : 0u;
  }
}

// Issue 7 async row fetches: each thread moves one 16B chunk of each 2KB row.
__device__ __forceinline__ void stage(float4 (*lbuf)[TPB],
                                      const uint32_t rows[NROW],
                                      const float* __restrict__ main_w,
                                      const float* __restrict__ shared_w,
                                      int tid) {
  const float4* r0 = (const float4*)(main_w + (size_t)rows[0] * DIM);
  async_ld_b128(&r0[tid], &lbuf[0][tid]);
#pragma unroll
  for (int r = 1; r < NROW; ++r) {
    const float4* rp = (const float4*)(shared_w + (size_t)rows[r] * DIM);
    async_ld_b128(&rp[tid], &lbuf[r][tid]);
  }
}

__global__ __launch_bounds__(TPB) void hash_ngram_embed_kernel(
    const int*   __restrict__ tokens,
    const float* __restrict__ main_w,
    const float* __restrict__ shared_w,
    const float* __restrict__ size_w,
    float*       __restrict__ out) {
  __shared__ float4 lbuf[2][NROW][TPB];  // 28 KB double buffer

  const int tid = threadIdx.x;
  const int p0  = blockIdx.x * PPB;

  // Per-thread chunk of sum_n size_w[n]  (tiny table, L2-resident)
  const float4* sw = (const float4*)size_w;
  float4 ssum = sw[tid];
#pragma unroll
  for (int n = 1; n < 6; ++n) {
    float4 v = sw[n * TPB + tid];
    ssum.x += v.x; ssum.y += v.y; ssum.z += v.z; ssum.w += v.w;
  }

  uint32_t rows[NROW];
  compute_rows(tokens, p0, rows);
  stage(lbuf[0], rows, main_w, shared_w, tid);   // ASYNCcnt = 7

  constexpr float inv7 = 1.0f / 7.0f;
#pragma unroll
  for (int i = 0; i < PPB; ++i) {
    const int cur = i & 1;
    if (i + 1 < PPB) {
      compute_rows(tokens, p0 + i + 1, rows);
      stage(lbuf[cur ^ 1], rows, main_w, shared_w, tid);  // ASYNCcnt <= 14
      wait_async<NROW>();   // async ops retire in order -> current buf ready
    } else {
      wait_async<0>();
    }
    __syncthreads();        // every wave waited on its own ASYNCcnt first

    float4 acc = lbuf[cur][0][tid];
#pragma unroll
    for (int r = 1; r < NROW; ++r) {
      float4 v = lbuf[cur][r][tid];
      acc.x += v.x; acc.y += v.y; acc.z += v.z; acc.w += v.w;
    }
    // Write-once output: non-temporal so it doesn't thrash L2 for the gathers
    v4f o;
    o.x = (acc.x + ssum.x) * inv7;
    o.y = (acc.y + ssum.y) * inv7;
    o.z = (acc.z + ssum.z) * inv7;
    o.w = (acc.w + ssum.w) * inv7;
    __builtin_nontemporal_store(o, (v4f*)out + (size_t)(p0 + i) * TPB + tid);

    __syncthreads();        // all reads of buf done before it is re-staged
  }
}

extern "C" void kernel_launch(void* const* d_in, const int* in_sizes, int n_in,
                              void* d_out, int out_size, void* d_ws, size_t ws_size,
                              hipStream_t stream) {
  (void)in_sizes; (void)n_in; (void)out_size; (void)d_ws; (void)ws_size;
  const int*   tokens   = (const int*)d_in[0];
  const float* main_w   = (const float*)d_in[1];
  const float* shared_w = (const float*)d_in[2];
  const float* size_w   = (const float*)d_in[3];
  float*       out      = (float*)d_out;

  const int nblocks = (BATCH * SEQ) / PPB;  // 4096
  hash_ngram_embed_kernel<<<dim3(nblocks), dim3(TPB), 0, stream>>>(
      tokens, main_w, shared_w, size_w, out);
}